// TreeTraversalDecisionTreeImplKeras_37744172597271
// MI455X (gfx1250) — compile-verified
//
#include <hip/hip_runtime.h>
#include <stdint.h>

// Problem constants (match reference)
#define NUM_TREES       128
#define NODES_PER_TREE  512
#define TOTAL_NODES     (NUM_TREES * NODES_PER_TREE)
#define NF              128   // features per sample
#define NC              4     // classes
#define DEPTH           8

// Tiling
#define BLOCK           128   // threads per block == samples per block
#define TPS             8     // trees staged per buffer (32 KB)
#define STAGES          (NUM_TREES / TPS)   // 16

// ---------------- CDNA5 async global->LDS copy helpers -----------------
// global_load_async_to_lds_b128: VDST = per-lane LDS byte offset (VGPR),
// VADDR = per-lane 64-bit global address. Tracked by ASYNCcnt.
__device__ __forceinline__ void async_copy_b128(uint32_t lds_byte_off, const void* gaddr) {
    asm volatile("global_load_async_to_lds_b128 %0, %1, off"
                 :: "v"(lds_byte_off), "v"(gaddr)
                 : "memory");
}

__device__ __forceinline__ void wait_asynccnt0() {
    asm volatile("s_wait_asynccnt 0" ::: "memory");
}

// Low 32 bits of a generic pointer to __shared__ == LDS byte offset
__device__ __forceinline__ uint32_t lds_offset(const void* p) {
    return (uint32_t)(uintptr_t)p;
}

// ---------------- Kernel 1: pack node tables -----------------
// meta = feat[7:0] | left[9b]<<8 | right[9b]<<17 ; pair with threshold bits.
// 8 bytes per node -> one ds_load_b64 per tree step.
__global__ void pack_nodes(const int* __restrict__ feat,
                           const float* __restrict__ thr,
                           const int* __restrict__ lft,
                           const int* __restrict__ rgt,
                           uint2* __restrict__ packed) {
    int i = blockIdx.x * blockDim.x + threadIdx.x;
    if (i < TOTAL_NODES) {
        uint32_t m = (uint32_t)(feat[i] & 255)
                   | ((uint32_t)(lft[i] & 511) << 8)
                   | ((uint32_t)(rgt[i] & 511) << 17);
        packed[i] = make_uint2(m, __float_as_uint(thr[i]));
    }
}

// ---------------- Kernel 2: forest traversal -----------------
__global__ __launch_bounds__(BLOCK) void forest_traverse(
    const float* __restrict__ x,        // [B, NF]
    const uint2* __restrict__ packed,   // [TOTAL_NODES] {meta, thr}
    const float* __restrict__ values,   // [TOTAL_NODES, NC]
    float* __restrict__ out)            // [B, NC]
{
    __shared__ float xs[BLOCK * NF];                     // 64 KB: x tile
    __shared__ uint2 ns[2 * TPS * NODES_PER_TREE];       // 2 x 32 KB: node buffers

    const int tid = threadIdx.x;
    const size_t sample0 = (size_t)blockIdx.x * BLOCK;

    // Per-stage async copy: TPS*512*8 B = 32 KB = 2048 x 16B -> 16 loads/thread
    constexpr int NODE_CHUNKS = (TPS * NODES_PER_TREE * 8) / 16 / BLOCK;  // 16
    auto stage_nodes = [&](int s, int buf) {
        const char* g = (const char*)(packed + (size_t)s * TPS * NODES_PER_TREE);
        const uint32_t l = lds_offset(&ns[buf * TPS * NODES_PER_TREE]);
        #pragma unroll
        for (int k = 0; k < NODE_CHUNKS; ++k) {
            int j = k * BLOCK + tid;
            async_copy_b128(l + j * 16, g + (size_t)j * 16);
        }
    };

    // ---- prologue: stage x tile (64 KB) + stage-0 nodes, all async ----
    {
        const char* g = (const char*)(x + sample0 * NF);
        const uint32_t l = lds_offset(&xs[0]);
        #pragma unroll
        for (int k = 0; k < (BLOCK * NF * 4) / 16 / BLOCK; ++k) {  // 32 iters
            int j = k * BLOCK + tid;
            async_copy_b128(l + j * 16, g + (size_t)j * 16);
        }
    }
    stage_nodes(0, 0);
    wait_asynccnt0();
    __syncthreads();

    float4 acc = make_float4(0.f, 0.f, 0.f, 0.f);
    const uint32_t rowBase = (uint32_t)tid * NF;

    for (int s = 0; s < STAGES; ++s) {
        // ---- prefetch next stage into the other buffer (overlapped) ----
        if (s + 1 < STAGES) stage_nodes(s + 1, (s + 1) & 1);

        const uint2* __restrict__ base = ns + (s & 1) * TPS * NODES_PER_TREE;

        // ---- traverse TPS trees with TPS-way ILP ----
        int node[TPS];
        #pragma unroll
        for (int t = 0; t < TPS; ++t) node[t] = 0;

        #pragma unroll
        for (int d = 0; d < DEPTH; ++d) {
            uint2 nd[TPS];
            #pragma unroll
            for (int t = 0; t < TPS; ++t)
                nd[t] = base[t * NODES_PER_TREE + node[t]];
            #pragma unroll
            for (int t = 0; t < TPS; ++t) {
                float fv = xs[rowBase + (nd[t].x & 255u)];
                node[t] = (fv < __uint_as_float(nd[t].y))
                            ? (int)((nd[t].x >> 8) & 511u)
                            : (int)((nd[t].x >> 17) & 511u);
            }
        }

        // ---- leaf gathers from L2-resident values table (16B aligned) ----
        #pragma unroll
        for (int t = 0; t < TPS; ++t) {
            const float4 v = *(const float4*)(
                values + ((size_t)(s * TPS + t) * NODES_PER_TREE + node[t]) * NC);
            acc.x += v.x; acc.y += v.y; acc.z += v.z; acc.w += v.w;
        }

        // prefetch issued at top of this iteration has had the whole stage
        // of compute to complete; wait is near-free in steady state.
        wait_asynccnt0();
        __syncthreads();
    }

    *(float4*)(out + (sample0 + tid) * NC) = acc;
}

// ---------------- Host launch -----------------
extern "C" void kernel_launch(void* const* d_in, const int* in_sizes, int n_in,
                              void* d_out, int out_size, void* d_ws, size_t ws_size,
                              hipStream_t stream) {
    const float* x          = (const float*)d_in[0];
    const int*   features   = (const int*)  d_in[1];
    const float* thresholds = (const float*)d_in[2];
    const int*   lefts      = (const int*)  d_in[3];
    const int*   rights     = (const int*)  d_in[4];
    const float* values     = (const float*)d_in[5];
    // d_in[6] = nodes_offset (t*512), recomputed implicitly.

    float* out    = (float*)d_out;
    uint2* packed = (uint2*)d_ws;        // TOTAL_NODES * 8 B = 512 KB scratch

    pack_nodes<<<(TOTAL_NODES + 255) / 256, 256, 0, stream>>>(
        features, thresholds, lefts, rights, packed);

    const int batch = in_sizes[0] / NF;  // 131072
    forest_traverse<<<batch / BLOCK, BLOCK, 0, stream>>>(x, packed, values, out);
}